// Autoencoder_40630390620397
// MI455X (gfx1250) — compile-verified
//
#include <hip/hip_runtime.h>
#include <math.h>

typedef __attribute__((ext_vector_type(2))) float v2f;
typedef __attribute__((ext_vector_type(8))) float v8f;

#define TT 512

// ---------------------------------------------------------------------------
// Kernel 1: fused encoder (conv1,conv2,conv3) + VQ softmax quantizer +
// decoder (dconv1,dconv2), one batch row per block, all activations in LDS.
// Writes flat[b, c*512+t] (B x 5120) to workspace for the FC GEMM.
// ---------------------------------------------------------------------------
__global__ __launch_bounds__(256) void enc_vq_dec(
    const float* __restrict__ x,
    const float* __restrict__ w1, const float* __restrict__ b1,
    const float* __restrict__ w2, const float* __restrict__ b2,
    const float* __restrict__ w3, const float* __restrict__ b3,
    const float* __restrict__ code,
    const float* __restrict__ d1w, const float* __restrict__ d1b,
    const float* __restrict__ d2w, const float* __restrict__ d2b,
    float* __restrict__ flat)
{
    // LDS layout (floats): weights + x row + double-buffered activation planes
    __shared__ float s[15674];   // 62,696 bytes
    const int W1 = 0,    B1 = 15,   W2 = 20,   B2 = 270,  W3 = 280, B3 = 980;
    const int CODE = 990, C2 = 1310, D1W = 1342, D1B = 1842, D2W = 1852, D2B = 2352;
    const int X = 2362, H1 = 2874, H2 = 5434, H3 = 10554;   // H2 reused for q, H3 for d1

    const int tid = threadIdx.x;
    const int b   = blockIdx.x;

    for (int i = tid; i < 15;  i += 256) s[W1  + i] = w1[i];
    for (int i = tid; i < 5;   i += 256) s[B1  + i] = b1[i];
    for (int i = tid; i < 250; i += 256) s[W2  + i] = w2[i];
    for (int i = tid; i < 10;  i += 256) s[B2  + i] = b2[i];
    for (int i = tid; i < 700; i += 256) s[W3  + i] = w3[i];
    for (int i = tid; i < 10;  i += 256) s[B3  + i] = b3[i];
    for (int i = tid; i < 320; i += 256) s[CODE + i] = code[i];
    for (int i = tid; i < 500; i += 256) s[D1W + i] = d1w[i];
    for (int i = tid; i < 10;  i += 256) s[D1B + i] = d1b[i];
    for (int i = tid; i < 500; i += 256) s[D2W + i] = d2w[i];
    for (int i = tid; i < 10;  i += 256) s[D2B + i] = d2b[i];
    for (int i = tid; i < TT;  i += 256) s[X + i] = x[b * TT + i];
    __syncthreads();

    // c2[k] = sum_c code[c][k]^2  (consumed only after later barriers)
    if (tid < 32) {
        float acc = 0.f;
        #pragma unroll
        for (int c = 0; c < 10; ++c) { float v = s[CODE + c * 32 + tid]; acc += v * v; }
        s[C2 + tid] = acc;
    }

    // conv1: (1 -> 5), k=3, pad=1, relu
    for (int idx = tid; idx < 5 * TT; idx += 256) {
        int c = idx >> 9, t = idx & (TT - 1);
        float acc = s[B1 + c];
        #pragma unroll
        for (int k = 0; k < 3; ++k) {
            int tt = t + k - 1;
            if (tt >= 0 && tt < TT) acc += s[W1 + c * 3 + k] * s[X + tt];
        }
        s[H1 + idx] = fmaxf(acc, 0.f);
    }
    __syncthreads();

    // conv2: (5 -> 10), k=5, pad=2, relu
    for (int idx = tid; idx < 10 * TT; idx += 256) {
        int c = idx >> 9, t = idx & (TT - 1);
        float acc = s[B2 + c];
        #pragma unroll
        for (int i = 0; i < 5; ++i)
            #pragma unroll
            for (int k = 0; k < 5; ++k) {
                int tt = t + k - 2;
                if (tt >= 0 && tt < TT) acc += s[W2 + (c * 5 + i) * 5 + k] * s[H1 + i * TT + tt];
            }
        s[H2 + idx] = fmaxf(acc, 0.f);
    }
    __syncthreads();

    // conv3: (10 -> 10), k=7, pad=3, tanh
    for (int idx = tid; idx < 10 * TT; idx += 256) {
        int c = idx >> 9, t = idx & (TT - 1);
        float acc = s[B3 + c];
        #pragma unroll
        for (int i = 0; i < 10; ++i)
            #pragma unroll
            for (int k = 0; k < 7; ++k) {
                int tt = t + k - 3;
                if (tt >= 0 && tt < TT) acc += s[W3 + (c * 10 + i) * 7 + k] * s[H2 + i * TT + tt];
            }
        s[H3 + idx] = tanhf(acc);
    }
    __syncthreads();

    // VQ: softmax(-dist) over 32 codes, q = prob @ code^T   (q written into H2)
    for (int t = tid; t < TT; t += 256) {
        float hv[10];
        #pragma unroll
        for (int c = 0; c < 10; ++c) hv[c] = s[H3 + c * TT + t];
        float x2 = 0.f;
        #pragma unroll
        for (int c = 0; c < 10; ++c) x2 += hv[c] * hv[c];

        float dist[32]; float dmin = 3.4e38f;
        #pragma unroll
        for (int k = 0; k < 32; ++k) {
            float cr = 0.f;
            #pragma unroll
            for (int c = 0; c < 10; ++c) cr += hv[c] * s[CODE + c * 32 + k];
            float d = x2 - 2.f * cr + s[C2 + k];
            dist[k] = d; dmin = fminf(dmin, d);
        }
        float S = 0.f, q[10];
        #pragma unroll
        for (int c = 0; c < 10; ++c) q[c] = 0.f;
        #pragma unroll
        for (int k = 0; k < 32; ++k) {
            float p = expf(dmin - dist[k]);
            S += p;
            #pragma unroll
            for (int c = 0; c < 10; ++c) q[c] += p * s[CODE + c * 32 + k];
        }
        float inv = 1.f / S;
        #pragma unroll
        for (int c = 0; c < 10; ++c) s[H2 + c * TT + t] = q[c] * inv;
    }
    __syncthreads();

    // dconv1: (10 -> 10), k=5, pad=2, relu   (H2 -> H3)
    for (int idx = tid; idx < 10 * TT; idx += 256) {
        int c = idx >> 9, t = idx & (TT - 1);
        float acc = s[D1B + c];
        #pragma unroll
        for (int i = 0; i < 10; ++i)
            #pragma unroll
            for (int k = 0; k < 5; ++k) {
                int tt = t + k - 2;
                if (tt >= 0 && tt < TT) acc += s[D1W + (c * 10 + i) * 5 + k] * s[H2 + i * TT + tt];
            }
        s[H3 + idx] = fmaxf(acc, 0.f);
    }
    __syncthreads();

    // dconv2: (10 -> 10), k=5, pad=2, relu -> flat[b, c*512 + t]
    for (int idx = tid; idx < 10 * TT; idx += 256) {
        int c = idx >> 9, t = idx & (TT - 1);
        float acc = s[D2B + c];
        #pragma unroll
        for (int i = 0; i < 10; ++i)
            #pragma unroll
            for (int k = 0; k < 5; ++k) {
                int tt = t + k - 2;
                if (tt >= 0 && tt < TT) acc += s[D2W + (c * 10 + i) * 5 + k] * s[H3 + i * TT + tt];
            }
        flat[b * 5120 + idx] = fmaxf(acc, 0.f);
    }
}

// ---------------------------------------------------------------------------
// Kernel 2: out = tanh(flat @ fcw^T + fcb) via V_WMMA_F32_16X16X4_F32.
// M=1024, N=512, K=5120.  Block: 8 waves = 128x64 tile; each wave 32x32
// (2x2 fragments, 2x reuse of each loaded fragment).
// A fragment (16x4 f32): lane l holds row (l&15), K = 2*(l>>4) + {0,1}.
// B fragment (4x16):     lane l holds col (l&15), K = 2*(l>>4) + {0,1}
//   -> B[k][n] = fcw[n][k], so the load addressing is identical to A's.
// ---------------------------------------------------------------------------
__global__ __launch_bounds__(256) void fc_wmma(
    const float* __restrict__ flat,   // 1024 x 5120
    const float* __restrict__ fcw,    // 512 x 5120
    const float* __restrict__ fcb,    // 512
    float* __restrict__ out)          // 1024 x 512
{
    const int lane = threadIdx.x & 31;
    const int wave = threadIdx.x >> 5;
    const int wm = wave & 3;          // 4 waves along M
    const int wn = wave >> 2;         // 2 waves along N
    const int m0 = blockIdx.x * 128 + wm * 32;
    const int n0 = blockIdx.y * 64  + wn * 32;

    const int row  = lane & 15;
    const int kh   = (lane >> 4) << 1;   // 0 or 2

    const float* a0p = flat + (size_t)(m0 + row)      * 5120 + kh;
    const float* a1p = flat + (size_t)(m0 + 16 + row) * 5120 + kh;
    const float* b0p = fcw  + (size_t)(n0 + row)      * 5120 + kh;
    const float* b1p = fcw  + (size_t)(n0 + 16 + row) * 5120 + kh;

    v8f c00 = {}, c01 = {}, c10 = {}, c11 = {};

    #pragma unroll 4
    for (int k = 0; k < 5120; k += 4) {
        v2f a0 = *(const v2f*)(a0p + k);
        v2f a1 = *(const v2f*)(a1p + k);
        v2f b0 = *(const v2f*)(b0p + k);
        v2f b1 = *(const v2f*)(b1p + k);
        c00 = __builtin_amdgcn_wmma_f32_16x16x4_f32(false, a0, false, b0, (short)0, c00, false, false);
        c01 = __builtin_amdgcn_wmma_f32_16x16x4_f32(false, a0, false, b1, (short)0, c01, false, false);
        c10 = __builtin_amdgcn_wmma_f32_16x16x4_f32(false, a1, false, b0, (short)0, c10, false, false);
        c11 = __builtin_amdgcn_wmma_f32_16x16x4_f32(false, a1, false, b1, (short)0, c11, false, false);
    }

    // C/D layout: v8f element r, lane l -> M = r + 8*(l>>4), N = (l&15)
    const int nA = n0 + (lane & 15);
    const int nB = nA + 16;
    const float biasA = fcb[nA];
    const float biasB = fcb[nB];
    const int mbase = m0 + 8 * (lane >> 4);
    #pragma unroll
    for (int r = 0; r < 8; ++r) {
        out[(size_t)(mbase + r)      * 512 + nA] = tanhf(c00[r] + biasA);
        out[(size_t)(mbase + r)      * 512 + nB] = tanhf(c01[r] + biasB);
        out[(size_t)(mbase + 16 + r) * 512 + nA] = tanhf(c10[r] + biasA);
        out[(size_t)(mbase + 16 + r) * 512 + nB] = tanhf(c11[r] + biasB);
    }
}

// ---------------------------------------------------------------------------
extern "C" void kernel_launch(void* const* d_in, const int* in_sizes, int n_in,
                              void* d_out, int out_size, void* d_ws, size_t ws_size,
                              hipStream_t stream) {
    const float* x    = (const float*)d_in[0];
    const float* w1   = (const float*)d_in[1];
    const float* b1   = (const float*)d_in[2];
    const float* w2   = (const float*)d_in[3];
    const float* b2   = (const float*)d_in[4];
    const float* w3   = (const float*)d_in[5];
    const float* b3   = (const float*)d_in[6];
    const float* code = (const float*)d_in[7];
    const float* d1w  = (const float*)d_in[8];
    const float* d1b  = (const float*)d_in[9];
    const float* d2w  = (const float*)d_in[10];
    const float* d2b  = (const float*)d_in[11];
    const float* fcw  = (const float*)d_in[12];
    const float* fcb  = (const float*)d_in[13];

    float* flat = (float*)d_ws;   // 1024 * 5120 floats = 20.97 MB

    enc_vq_dec<<<1024, 256, 0, stream>>>(x, w1, b1, w2, b2, w3, b3, code,
                                         d1w, d1b, d2w, d2b, flat);
    fc_wmma<<<dim3(8, 8), 256, 0, stream>>>(flat, fcw, fcb, (float*)d_out);
}